// CnnTextClassifier_18494129176705
// MI455X (gfx1250) — compile-verified
//
#include <hip/hip_runtime.h>
#include <math.h>

typedef __attribute__((ext_vector_type(2))) float v2f;
typedef __attribute__((ext_vector_type(8))) float v8f;

#define LQ   512
#define EMB  100
#define DP   113   // LDS row stride for Q/K tiles (odd -> bank-conflict free)
#define XPAD 112   // row stride of intermediate x in workspace (16B-aligned rows)

// gfx1250 has a native v_tanh_f32 transcendental op; use it if clang exposes it.
#if __has_builtin(__builtin_amdgcn_tanhf)
#define FAST_TANH(x) __builtin_amdgcn_tanhf(x)
#else
#define FAST_TANH(x) tanhf(x)
#endif

// ---------------------------------------------------------------------------
// Kernel 1: fused triple attention + residual.
// One wave32 per (batch, 16-query-row tile). grid = B * (L/16) = 128 blocks.
// ---------------------------------------------------------------------------
__global__ __launch_bounds__(32) void attn_x_kernel(const float* __restrict__ src,
                                                    const float* __restrict__ att_w,
                                                    float* __restrict__ xout) {
    __shared__ float Qs[16][DP];
    __shared__ float Ks[16][DP];
    __shared__ float S1[16][LQ];   // dot-product scores, later combined probs
    __shared__ float S2[16][LQ];   // additive (tanh) scores

    const int lane = threadIdx.x;          // 0..31
    const int b    = blockIdx.x >> 5;      // / (L/16)
    const int qt   = blockIdx.x & 31;      // query tile
    const int half = lane >> 4;            // 0/1
    const int lm   = lane & 15;            // lane % 16
    const float* srcB = src + (size_t)b * LQ * EMB;

    // load Q tile (zero-padded in d)
    for (int i = lane; i < 16 * DP; i += 32) {
        int m = i / DP, d = i % DP;
        Qs[m][d] = (d < EMB) ? srcB[(qt * 16 + m) * EMB + d] : 0.0f;
    }
    __syncthreads();

    // ---- pass 1: score tiles -------------------------------------------------
    for (int kt = 0; kt < LQ / 16; ++kt) {
        for (int i = lane; i < 16 * DP; i += 32) {
            int m = i / DP, d = i % DP;
            Ks[m][d] = (d < EMB) ? srcB[(kt * 16 + m) * EMB + d] : 0.0f;
        }
        __syncthreads();

        // S1 tile = Q (16xd) * K^T (dx16) via V_WMMA_F32_16X16X4_F32
        v8f c = {0.0f, 0.0f, 0.0f, 0.0f, 0.0f, 0.0f, 0.0f, 0.0f};
        for (int d0 = 0; d0 < 112; d0 += 4) {
            int kd = d0 + 2 * half;        // A/B: vgpr v holds kd = v + 2*half
            v2f a, bb;
            a[0]  = Qs[lm][kd];  a[1]  = Qs[lm][kd + 1];
            bb[0] = Ks[lm][kd];  bb[1] = Ks[lm][kd + 1];
            c = __builtin_amdgcn_wmma_f32_16x16x4_f32(false, a, false, bb,
                                                      (short)0, c, false, false);
        }
#pragma unroll
        for (int v = 0; v < 8; ++v)        // D: m = v + 8*half, n = lm
            S1[v + 8 * half][kt * 16 + lm] = c[v];

        // additive scores: s2[m][n] = sum_d tanh(Q[m][d] + K[n][d])
        // d outer so the K value is loaded once and shared by the 8 row accums.
        {
            float acc[8] = {0.f, 0.f, 0.f, 0.f, 0.f, 0.f, 0.f, 0.f};
            const float* qrow = &Qs[8 * half][0];
            for (int d = 0; d < EMB; ++d) {
                float kv = Ks[lm][d];
#pragma unroll
                for (int v = 0; v < 8; ++v)
                    acc[v] += FAST_TANH(qrow[v * DP + d] + kv);
            }
#pragma unroll
            for (int v = 0; v < 8; ++v)
                S2[v + 8 * half][kt * 16 + lm] = acc[v];
        }
        __syncthreads();
    }

    // ---- pass 2: three softmaxes fused into combined probability matrix -----
    float aw0 = att_w[0], aw1 = att_w[1], aw2 = att_w[2];
    float winv = 1.0f / (aw0 + aw1 + aw2);
    float w0 = aw0 * winv, w1 = aw1 * winv, w2 = aw2 * winv;
    const float sc3 = 0.1f;  // 1/sqrt(EMB)

    for (int m = 0; m < 16; ++m) {
        float m1 = -1e30f, m2 = -1e30f;
        for (int k = lane; k < LQ; k += 32) {
            m1 = fmaxf(m1, S1[m][k]);
            m2 = fmaxf(m2, S2[m][k]);
        }
        for (int off = 16; off > 0; off >>= 1) {
            m1 = fmaxf(m1, __shfl_xor(m1, off));
            m2 = fmaxf(m2, __shfl_xor(m2, off));
        }
        float m3 = m1 * sc3;               // x0.1 is monotone -> same argmax
        float z1 = 0.0f, z2 = 0.0f, z3 = 0.0f;
        for (int k = lane; k < LQ; k += 32) {
            float s1v = S1[m][k];
            z1 += __expf(s1v - m1);
            z2 += __expf(S2[m][k] - m2);
            z3 += __expf(s1v * sc3 - m3);
        }
        for (int off = 16; off > 0; off >>= 1) {
            z1 += __shfl_xor(z1, off);
            z2 += __shfl_xor(z2, off);
            z3 += __shfl_xor(z3, off);
        }
        float r1 = w0 / z1, r2 = w1 / z2, r3 = w2 / z3;
        for (int k = lane; k < LQ; k += 32) {
            float s1v = S1[m][k];
            S1[m][k] = r1 * __expf(s1v - m1)
                     + r2 * __expf(S2[m][k] - m2)
                     + r3 * __expf(s1v * sc3 - m3);
        }
    }
    __syncthreads();

    // ---- pass 3: x = src + P * V via WMMA ------------------------------------
    for (int d0 = 0; d0 < XPAD; d0 += 16) {
        int dcol = d0 + lm;
        bool dok = dcol < EMB;
        v8f c = {0.0f, 0.0f, 0.0f, 0.0f, 0.0f, 0.0f, 0.0f, 0.0f};
        for (int k0 = 0; k0 < LQ; k0 += 4) {
            int kd = k0 + 2 * half;
            v2f a, bb;
            a[0]  = S1[lm][kd];  a[1] = S1[lm][kd + 1];
            bb[0] = dok ? srcB[kd * EMB + dcol] : 0.0f;
            bb[1] = dok ? srcB[(kd + 1) * EMB + dcol] : 0.0f;
            c = __builtin_amdgcn_wmma_f32_16x16x4_f32(false, a, false, bb,
                                                      (short)0, c, false, false);
        }
#pragma unroll
        for (int v = 0; v < 8; ++v) {
            int q = qt * 16 + v + 8 * half;
            float val = dok ? (c[v] + srcB[q * EMB + dcol]) : 0.0f;
            xout[((size_t)b * LQ + q) * XPAD + dcol] = val;
        }
    }
}

// ---------------------------------------------------------------------------
// Kernel 2: conv branches (ws=1,2,3, pad ws-1 both sides) + ReLU + max-pool
// + final linear. One block per batch.
// ---------------------------------------------------------------------------
__global__ __launch_bounds__(256) void head_kernel(const float* __restrict__ x,
    const float* __restrict__ w1, const float* __restrict__ b1,
    const float* __restrict__ w2, const float* __restrict__ b2,
    const float* __restrict__ w3, const float* __restrict__ b3,
    const float* __restrict__ lw, const float* __restrict__ lb,
    float* __restrict__ out) {
    __shared__ float pooled[30];
    __shared__ float red[256];
    const int b = blockIdx.x, tid = threadIdx.x;
    const float* xb = x + (size_t)b * LQ * XPAD;

    for (int ch = 0; ch < 30; ++ch) {
        const int br = ch / 10, oc = ch % 10, ws = br + 1;
        const float* wgt = (br == 0 ? w1 : (br == 1 ? w2 : w3)) + oc * ws * EMB;
        const float bias = (br == 0 ? b1 : (br == 1 ? b2 : b3))[oc];
        const int TL = LQ + ws - 1;       // output length with pad (ws-1, ws-1)
        float local = 0.0f;               // ReLU folded into max init
        for (int t = tid; t < TL; t += 256) {
            float acc = bias;
            for (int i = 0; i < ws; ++i) {
                int r = t + i - (ws - 1);
                if (r >= 0 && r < LQ) {
                    const float4* xr = (const float4*)(xb + r * XPAD);
                    const float4* wr = (const float4*)(wgt + i * EMB);
                    for (int d = 0; d < EMB / 4; ++d) {
                        float4 xv = xr[d], wv = wr[d];
                        acc += xv.x * wv.x + xv.y * wv.y + xv.z * wv.z + xv.w * wv.w;
                    }
                }
            }
            local = fmaxf(local, acc);
        }
        red[tid] = local;
        __syncthreads();
        for (int s = 128; s > 0; s >>= 1) {
            if (tid < s) red[tid] = fmaxf(red[tid], red[tid + s]);
            __syncthreads();
        }
        if (tid == 0) pooled[ch] = red[0];
        __syncthreads();
    }
    if (tid < 2) {
        float acc = lb[tid];
        for (int c = 0; c < 30; ++c) acc += pooled[c] * lw[c * 2 + tid];
        out[b * 2 + tid] = acc;
    }
}

extern "C" void kernel_launch(void* const* d_in, const int* in_sizes, int n_in,
                              void* d_out, int out_size, void* d_ws, size_t ws_size,
                              hipStream_t stream) {
    (void)in_sizes; (void)n_in; (void)out_size; (void)ws_size;
    const float* src   = (const float*)d_in[0];
    const float* att_w = (const float*)d_in[1];
    const float* w1    = (const float*)d_in[2];
    const float* b1    = (const float*)d_in[3];
    const float* w2    = (const float*)d_in[4];
    const float* b2    = (const float*)d_in[5];
    const float* w3    = (const float*)d_in[6];
    const float* b3    = (const float*)d_in[7];
    const float* lw    = (const float*)d_in[8];
    const float* lb    = (const float*)d_in[9];
    float* xout = (float*)d_ws;   // [4][512][XPAD]

    attn_x_kernel<<<4 * (LQ / 16), 32, 0, stream>>>(src, att_w, xout);
    head_kernel<<<4, 256, 0, stream>>>(xout, w1, b1, w2, b2, w3, b3, lw, lb,
                                       (float*)d_out);
}